// GradientLoss_10668698763438
// MI455X (gfx1250) — compile-verified
//
#include <hip/hip_runtime.h>
#include <math.h>

typedef __attribute__((ext_vector_type(2))) float v2f;
typedef __attribute__((ext_vector_type(8))) float v8f;

#define IMG_H 512
#define IMG_W 512
#define N_IMG 32
#define TS 32              // output tile: 32x32
#define HALO 34            // tile + 1-pixel halo
#define LPITCH 36          // LDS pitch (pad to dodge bank conflicts)
#define NQ 7               // number of reduced quantities per image

__global__ void zero_accum_kernel(float* __restrict__ acc, int n) {
    int i = blockIdx.x * blockDim.x + threadIdx.x;
    if (i < n) acc[i] = 0.0f;
}

__global__ __launch_bounds__(256) void gradloss_tile_kernel(
    const float* __restrict__ pred,
    const float* __restrict__ targ,
    float* __restrict__ accum)   // [N_IMG][8]: mask, mag, dir, lp, lp2, lt, lt2
{
    __shared__ float sp[HALO * LPITCH];   // sigmoid(pred) tile + halo (zero-padded)
    __shared__ float st[HALO * LPITCH];   // target tile + halo (zero-padded)
    __shared__ float red[NQ * 256];       // per-thread partials for block reduction

    const int tid = threadIdx.x;
    const int ox  = blockIdx.x * TS;
    const int oy  = blockIdx.y * TS;
    const int img = blockIdx.z;
    const size_t base = (size_t)img * (IMG_H * IMG_W);
    const float* __restrict__ P = pred + base;
    const float* __restrict__ T = targ + base;

    // ---- cooperative halo load (zero pad outside image; sigmoid fused on pred) ----
    for (int idx = tid; idx < HALO * HALO; idx += 256) {
        int r  = idx / HALO;
        int c  = idx - r * HALO;
        int gy = oy - 1 + r;
        int gx = ox - 1 + c;
        float pv = 0.0f, tv = 0.0f;
        if (gy >= 0 && gy < IMG_H && gx >= 0 && gx < IMG_W) {
            int off = gy * IMG_W + gx;
            float raw = P[off];
            pv = 1.0f / (1.0f + expf(-raw));   // sigmoid
            tv = T[off];
        }
        sp[r * LPITCH + c] = pv;
        st[r * LPITCH + c] = tv;
    }
    __syncthreads();

    // ---- per-thread stencil compute over 4 pixels ----
    float a0 = 0.f, a1 = 0.f, a2 = 0.f, a3 = 0.f, a4 = 0.f, a5 = 0.f, a6 = 0.f;
#pragma unroll
    for (int j = 0; j < 4; ++j) {
        int p  = tid + 256 * j;           // 0..1023
        int py = p >> 5;
        int px = p & 31;
        const float* sP = &sp[(py + 1) * LPITCH + (px + 1)];
        const float* sT = &st[(py + 1) * LPITCH + (px + 1)];

        float p_mm = sP[-LPITCH - 1], p_m0 = sP[-LPITCH], p_mp = sP[-LPITCH + 1];
        float p_0m = sP[-1],          p_00 = sP[0],       p_0p = sP[1];
        float p_pm = sP[ LPITCH - 1], p_p0 = sP[ LPITCH], p_pp = sP[ LPITCH + 1];

        float t_mm = sT[-LPITCH - 1], t_m0 = sT[-LPITCH], t_mp = sT[-LPITCH + 1];
        float t_0m = sT[-1],          t_00 = sT[0],       t_0p = sT[1];
        float t_pm = sT[ LPITCH - 1], t_p0 = sT[ LPITCH], t_pp = sT[ LPITCH + 1];

        // Sobel (cross-correlation, zero-padded SAME)
        float pgx = (p_mp + 2.0f * p_0p + p_pp) - (p_mm + 2.0f * p_0m + p_pm);
        float pgy = (p_pm + 2.0f * p_p0 + p_pp) - (p_mm + 2.0f * p_m0 + p_mp);
        float tgx = (t_mp + 2.0f * t_0p + t_pp) - (t_mm + 2.0f * t_0m + t_pm);
        float tgy = (t_pm + 2.0f * t_p0 + t_pp) - (t_mm + 2.0f * t_m0 + t_mp);

        float pmag = sqrtf(pgx * pgx + pgy * pgy + 1e-6f);
        float tmag = sqrtf(tgx * tgx + tgy * tgy + 1e-6f);

        float dm   = pmag - tmag;
        float mask = (tmag > 0.1f) ? 1.0f : 0.0f;

        float invp = 1.0f / (pmag + 1e-6f);
        float invt = 1.0f / (tmag + 1e-6f);
        float dot  = (pgx * tgx + pgy * tgy) * invp * invt;
        float dl   = fminf(fmaxf(1.0f - dot, 0.0f), 2.0f);

        // Laplacian
        float lp = p_m0 + p_p0 + p_0m + p_0p - 4.0f * p_00;
        float lt = t_m0 + t_p0 + t_0m + t_0p - 4.0f * t_00;

        a0 += mask;
        a1 += dm * dm * mask;
        a2 += dl * mask;
        a3 += lp;
        a4 += lp * lp;
        a5 += lt;
        a6 += lt * lt;
    }

    red[0 * 256 + tid] = a0;
    red[1 * 256 + tid] = a1;
    red[2 * 256 + tid] = a2;
    red[3 * 256 + tid] = a3;
    red[4 * 256 + tid] = a4;
    red[5 * 256 + tid] = a5;
    red[6 * 256 + tid] = a6;
    __syncthreads();

    // ---- block reduction on the matrix pipe ----
    // Wave 0 packs 256 partials into a 16x4 f32 A-matrix (2 VGPRs/lane, documented
    // layout) and multiplies by an all-ones B (layout-independent): D rows carry
    // row-sums; lane0 rows 0-7, lane16 rows 8-15 -> one shfl_down(16) completes it.
    if (tid < 32) {
        const int lane = tid;
        for (int q = 0; q < NQ; ++q) {
            const float* src = &red[q * 256];
            float s0 = 0.f, s1 = 0.f;
#pragma unroll
            for (int k = 0; k < 4; ++k) s0 += src[lane * 8 + k];
#pragma unroll
            for (int k = 4; k < 8; ++k) s1 += src[lane * 8 + k];
            v2f a; a[0] = s0; a[1] = s1;
            v2f b; b[0] = 1.0f; b[1] = 1.0f;
            v8f c = {};
            v8f d = __builtin_amdgcn_wmma_f32_16x16x4_f32(
                false, a, false, b, (short)0, c, false, false);
            float s = d[0] + d[1] + d[2] + d[3] + d[4] + d[5] + d[6] + d[7];
            s += __shfl_down(s, 16, 32);            // lane0 += lane16 (rows 8..15)
            if (lane == 0) atomicAdd(&accum[img * 8 + q], s);
        }
    }
}

__global__ void gradloss_finalize_kernel(const float* __restrict__ accum,
                                         float* __restrict__ out)
{
    const int i = threadIdx.x;           // one lane per image, 32 lanes = 1 wave
    const float N   = (float)(IMG_H * IMG_W);
    const float Nm1 = N - 1.0f;

    float mask = accum[i * 8 + 0];
    float mag  = accum[i * 8 + 1];
    float dir  = accum[i * 8 + 2];
    float lp   = accum[i * 8 + 3];
    float lp2  = accum[i * 8 + 4];
    float lt   = accum[i * 8 + 5];
    float lt2  = accum[i * 8 + 6];

    float valid = mask + 1e-6f;
    float magn  = mag / valid;
    float dirn  = dir / valid;
    float varp  = (lp2 - lp * lp / N) / Nm1;   // unbiased (ddof=1)
    float vart  = (lt2 - lt * lt / N) / Nm1;
    float dv    = varp - vart;
    float sh    = dv * dv;

#pragma unroll
    for (int off = 16; off > 0; off >>= 1) {
        magn += __shfl_down(magn, off, 32);
        dirn += __shfl_down(dirn, off, 32);
        sh   += __shfl_down(sh,   off, 32);
    }
    if (i == 0) {
        const float inv = 1.0f / (float)N_IMG;
        out[0] = 0.6f * (magn * inv) + 0.4f * (dirn * inv) + 1.2f * (sh * inv);
    }
}

extern "C" void kernel_launch(void* const* d_in, const int* in_sizes, int n_in,
                              void* d_out, int out_size, void* d_ws, size_t ws_size,
                              hipStream_t stream) {
    const float* pred = (const float*)d_in[0];
    const float* targ = (const float*)d_in[1];
    float* out   = (float*)d_out;
    float* accum = (float*)d_ws;                 // N_IMG * 8 floats

    zero_accum_kernel<<<1, 256, 0, stream>>>(accum, N_IMG * 8);

    dim3 grid(IMG_W / TS, IMG_H / TS, N_IMG);    // 16 x 16 x 32
    gradloss_tile_kernel<<<grid, 256, 0, stream>>>(pred, targ, accum);

    gradloss_finalize_kernel<<<1, 32, 0, stream>>>(accum, out);
}